// LIF0Neuron_61409442398456
// MI455X (gfx1250) — compile-verified
//
#include <hip/hip_runtime.h>
#include <stdint.h>

// Problem constants (from reference): x is (T=16, B=16, C=128, H=32, W=32) fp32.
#define T_STEPS 16
#define B_DIM   16
#define N_PER   (128 * 32 * 32)   // 131072 elements per (t,b) sample
#define K_SEL   (N_PER / 2)       // k = N * 50 / 100 = 65536
#define BETA_F  0.25f
#define CHUNK   4096              // elements processed per block in streaming kernels
#define BLOCKS_PER_ROW (N_PER / CHUNK)  // 32
#define NBINS   2048

// Native Clang vector (not HIP_vector_type struct) so nontemporal builtins accept it.
typedef float v4f __attribute__((ext_vector_type(4)));

// Order-preserving float -> uint32 key: key(a) >= key(b)  <=>  a >= b (finite floats).
__device__ __forceinline__ uint32_t okey(float f) {
    uint32_t u = __float_as_uint(f);
    return u ^ ((uint32_t)((int32_t)u >> 31) | 0x80000000u);
}

// ---------------------------------------------------------------------------
// Kernel 1: fused spike(t-1) emit + reset + leaky integrate + histogram pass 0
// Grid: B*BLOCKS_PER_ROW blocks x 256 threads (8 wave32 waves).
// x is read once (non-temporal), spikes written once (non-temporal) so the
// reused 8MB membrane + histograms stay resident in the 192MB L2.
// ---------------------------------------------------------------------------
__global__ void lif_update_hist(const float* __restrict__ xt,
                                float* __restrict__ mem,
                                uint32_t* __restrict__ hist,
                                const uint32_t* __restrict__ sel,
                                float* __restrict__ out_prev,
                                int t) {
    __shared__ uint32_t lhist[NBINS];
    const int tid   = threadIdx.x;
    const int b     = blockIdx.x / BLOCKS_PER_ROW;
    const int chunk = blockIdx.x % BLOCKS_PER_ROW;
    const int base4 = (b * N_PER + chunk * CHUNK) / 4;  // v4f index base

#pragma unroll
    for (int i = 0; i < NBINS / 256; ++i) lhist[tid + i * 256] = 0;  // straight-line ds stores
    uint32_t thr = 0u;
    if (t > 0) thr = sel[2 * b];   // threshold key from previous round
    __syncthreads();

    const v4f* mem4r = (const v4f*)mem;
    const v4f* x4    = (const v4f*)xt;
    v4f*       mem4w = (v4f*)mem;
    v4f*       out4  = (v4f*)out_prev;

#pragma unroll
    for (int it = 0; it < 4; ++it) {
        const int e4 = base4 + it * 256 + tid;
        // gfx1250 prefetch of the next x tile (speculative; OOB silently dropped)
        __builtin_prefetch(&x4[e4 + 256], 0, 0);

        v4f m = mem4r[e4];
        if (t > 0) {
            v4f s;
            s.x = (okey(m.x) >= thr) ? 1.f : 0.f;  m.x = (s.x != 0.f) ? 0.f : m.x;
            s.y = (okey(m.y) >= thr) ? 1.f : 0.f;  m.y = (s.y != 0.f) ? 0.f : m.y;
            s.z = (okey(m.z) >= thr) ? 1.f : 0.f;  m.z = (s.z != 0.f) ? 0.f : m.z;
            s.w = (okey(m.w) >= thr) ? 1.f : 0.f;  m.w = (s.w != 0.f) ? 0.f : m.w;
            __builtin_nontemporal_store(s, &out4[e4]);  // spikes for step t-1, write-once
        }
        const v4f xv = __builtin_nontemporal_load(&x4[e4]);  // read-once stream
        m = BETA_F * m + xv;     // vector FMA
        mem4w[e4] = m;           // regular TH: re-read 2x this step -> keep in L2

        atomicAdd(&lhist[okey(m.x) >> 21], 1u);
        atomicAdd(&lhist[okey(m.y) >> 21], 1u);
        atomicAdd(&lhist[okey(m.z) >> 21], 1u);
        atomicAdd(&lhist[okey(m.w) >> 21], 1u);
    }
    __syncthreads();
#pragma unroll
    for (int i = 0; i < NBINS / 256; ++i) {
        const uint32_t v = lhist[tid + i * 256];
        if (v) atomicAdd(&hist[b * NBINS + tid + i * 256], v);
    }
}

// ---------------------------------------------------------------------------
// Kernel 2: per-row descending scan of histogram -> fix bits, carry residual
// rank, and zero the histogram for the next pass. One block per batch row.
// ---------------------------------------------------------------------------
__global__ void lif_select(uint32_t* __restrict__ hist,
                           uint32_t* __restrict__ sel,
                           int pass) {
    __shared__ uint32_t h[NBINS];
    const int b   = blockIdx.x;
    const int tid = threadIdx.x;
    const int nb  = (pass == 2) ? 1024 : NBINS;
    for (int i = tid; i < nb; i += 256) {
        h[i] = hist[b * NBINS + i];
        hist[b * NBINS + i] = 0;  // ready for next histogram pass
    }
    __syncthreads();
    if (tid == 0) {
        uint32_t r      = (pass == 0) ? (uint32_t)K_SEL : sel[2 * b + 1];
        uint32_t prefix = (pass == 0) ? 0u : sel[2 * b];
        int bin = nb - 1;
        for (; bin > 0; --bin) {
            const uint32_t c = h[bin];
            if (c >= r) break;
            r -= c;
        }
        if (pass == 0)      prefix  = (uint32_t)bin << 21;
        else if (pass == 1) prefix |= (uint32_t)bin << 10;
        else                prefix |= (uint32_t)bin;
        sel[2 * b]     = prefix;   // after pass 2 this IS the threshold key
        sel[2 * b + 1] = r;
    }
}

// ---------------------------------------------------------------------------
// Kernel 3: refine histogram pass. Stages the block's 16KB tile into LDS with
// CDNA5 async global->LDS copies (ASYNCcnt); each lane consumes only its own
// 16B slots, so a per-wave s_wait_asynccnt suffices before the LDS reads.
// ---------------------------------------------------------------------------
__global__ void lif_hist_refine(const float* __restrict__ mem,
                                uint32_t* __restrict__ hist,
                                const uint32_t* __restrict__ sel,
                                int matchShift, int binShift, uint32_t binMask) {
    __shared__ __align__(16) uint32_t lds[NBINS + CHUNK];  // 8KB hist + 16KB tile
    const int tid   = threadIdx.x;
    const int b     = blockIdx.x / BLOCKS_PER_ROW;
    const int chunk = blockIdx.x % BLOCKS_PER_ROW;
    const int base  = b * N_PER + chunk * CHUNK;

#pragma unroll
    for (int i = 0; i < NBINS / 256; ++i) lds[tid + i * 256] = 0;
    const uint32_t match = sel[2 * b] >> matchShift;

    // Issue 4 async b128 loads per lane: global membrane -> LDS tile slots.
#pragma unroll
    for (int it = 0; it < 4; ++it) {
        const float*   g        = mem + base + it * 1024 + tid * 4;
        const uint32_t lds_byte =
            (uint32_t)(uintptr_t)(&lds[NBINS + it * 1024 + tid * 4]);
        asm volatile("global_load_async_to_lds_b128 %0, %1, off"
                     :: "v"(lds_byte), "v"(g) : "memory");
    }
    asm volatile("s_wait_asynccnt 0" ::: "memory");  // own slots complete
    __syncthreads();                                  // lhist zeroing visible

#pragma unroll
    for (int it = 0; it < 4; ++it) {
#pragma unroll
        for (int j = 0; j < 4; ++j) {
            const float    v   = __uint_as_float(lds[NBINS + it * 1024 + tid * 4 + j]);
            const uint32_t key = okey(v);
            if ((key >> matchShift) == match)
                atomicAdd(&lds[(key >> binShift) & binMask], 1u);
        }
    }
    __syncthreads();
#pragma unroll
    for (int i = 0; i < NBINS / 256; ++i) {
        const uint32_t v = lds[tid + i * 256];
        if (v) atomicAdd(&hist[b * NBINS + tid + i * 256], v);
    }
}

// ---------------------------------------------------------------------------
// Kernel 4: emit spikes for the final time step (t = T-1).
// ---------------------------------------------------------------------------
__global__ void lif_spike_last(const float* __restrict__ mem,
                               const uint32_t* __restrict__ sel,
                               float* __restrict__ out_last) {
    const int e4 = blockIdx.x * 256 + threadIdx.x;   // v4f index
    const int b  = (e4 * 4) / N_PER;
    const uint32_t thr = sel[2 * b];
    const v4f m = ((const v4f*)mem)[e4];
    v4f s;
    s.x = (okey(m.x) >= thr) ? 1.f : 0.f;
    s.y = (okey(m.y) >= thr) ? 1.f : 0.f;
    s.z = (okey(m.z) >= thr) ? 1.f : 0.f;
    s.w = (okey(m.w) >= thr) ? 1.f : 0.f;
    __builtin_nontemporal_store(s, &((v4f*)out_last)[e4]);
}

// ---------------------------------------------------------------------------
extern "C" void kernel_launch(void* const* d_in, const int* in_sizes, int n_in,
                              void* d_out, int out_size, void* d_ws, size_t ws_size,
                              hipStream_t stream) {
    const float* x   = (const float*)d_in[0];
    float*       out = (float*)d_out;
    const size_t BN  = (size_t)B_DIM * (size_t)N_PER;  // elements per time step

    // Workspace layout: [membrane: BN floats][hist: B*2048 u32][sel: B*2 u32]
    float*    mem  = (float*)d_ws;
    uint32_t* hist = (uint32_t*)((char*)d_ws + BN * sizeof(float));
    uint32_t* sel  = hist + (size_t)B_DIM * NBINS;
    const size_t clear_bytes =
        BN * sizeof(float) + (size_t)B_DIM * NBINS * 4 + (size_t)B_DIM * 2 * 4;
    (void)hipMemsetAsync(d_ws, 0, clear_bytes, stream);  // capture-safe memset node

    const int stream_blocks = B_DIM * BLOCKS_PER_ROW;  // 512
    for (int t = 0; t < T_STEPS; ++t) {
        const float* xt       = x + (size_t)t * BN;
        float*       out_prev = out + (size_t)((t > 0) ? (t - 1) : 0) * BN;

        lif_update_hist<<<stream_blocks, 256, 0, stream>>>(xt, mem, hist, sel,
                                                           out_prev, t);
        lif_select<<<B_DIM, 256, 0, stream>>>(hist, sel, 0);
        lif_hist_refine<<<stream_blocks, 256, 0, stream>>>(mem, hist, sel,
                                                           21, 10, 2047u);
        lif_select<<<B_DIM, 256, 0, stream>>>(hist, sel, 1);
        lif_hist_refine<<<stream_blocks, 256, 0, stream>>>(mem, hist, sel,
                                                           10, 0, 1023u);
        lif_select<<<B_DIM, 256, 0, stream>>>(hist, sel, 2);
    }
    lif_spike_last<<<(int)(BN / 4 / 256), 256, 0, stream>>>(
        mem, sel, out + (size_t)(T_STEPS - 1) * BN);
}